// Chebyshev_85478439125126
// MI455X (gfx1250) — compile-verified
//
#include <hip/hip_runtime.h>
#include <stdint.h>

// ---- problem constants (match reference) ----
#define B_    8
#define M_    49152
#define FIN_  32
#define FOUT_ 64
#define K_    5
#define DEG_  9
#define C_    (FIN_ * B_)    // 256 columns of the Chebyshev basis matrices
#define KD_   (K_ * FIN_)    // 160 = contraction length of the dense step

#define SW_FLOATS (KD_ * FOUT_)        // 10240 floats = 40960 B of kernel weights
#define SA_FLOATS (K_ * 16 * C_)       // 20480 floats = 81920 B slab of xs
#define SA_LDS_OFF (SW_FLOATS * 4)     // byte offset of the A slab in LDS (40960)

typedef __attribute__((ext_vector_type(2))) float v2f;
typedef __attribute__((ext_vector_type(8))) float v8f;
typedef __attribute__((ext_vector_type(4))) unsigned int u32x4;
typedef __attribute__((ext_vector_type(8))) int i32x8;
typedef __attribute__((ext_vector_type(4))) int i32x4;

// ---------------------------------------------------------------------------
// Kernel 1: x0[m, f*B + b] = x[b, m, f]   (coalesced writes over x0)
// ---------------------------------------------------------------------------
__global__ __launch_bounds__(256) void cheb_permute(const float* __restrict__ x,
                                                    float* __restrict__ x0) {
    int idx = blockIdx.x * blockDim.x + threadIdx.x;   // over M_*C_
    int c = idx & (C_ - 1);
    int m = idx >> 8;                                  // C_ == 256
    int f = c >> 3;                                    // B_ == 8
    int b = c & (B_ - 1);
    x0[idx] = x[((size_t)b * M_ + m) * FIN_ + f];
}

// ---------------------------------------------------------------------------
// Kernel 2: y = alpha * (L @ xin) + beta * xprev
// L is COO with exactly DEG_ entries per row, rows sorted -> row id implicit.
// One 256-thread block per row; each thread owns one of the 256 columns.
// Neighbor-row reads are fully coalesced (256 consecutive floats per nnz).
// ---------------------------------------------------------------------------
__global__ __launch_bounds__(256) void cheb_spmm(const int* __restrict__ cols,
                                                 const float* __restrict__ vals,
                                                 const float* __restrict__ xin,
                                                 const float* __restrict__ xprev,
                                                 float* __restrict__ y,
                                                 float alpha, float beta) {
    int row = blockIdx.x;
    int c = threadIdx.x;
    const int*   rc = cols + (size_t)row * DEG_;
    const float* rv = vals + (size_t)row * DEG_;

    float acc = 0.0f;
#pragma unroll
    for (int j = 0; j < DEG_; ++j) {
        acc += rv[j] * xin[(size_t)rc[j] * C_ + c];
    }
    float prev = xprev[(size_t)row * C_ + c];
    y[(size_t)row * C_ + c] = alpha * acc + beta * prev;
}

// ---------------------------------------------------------------------------
// Kernel 3: dense contraction with V_WMMA_F32_16X16X4_F32 + TDM staging.
//   out[b, m, o] = sum_j A[m, j] * W[j, o],   j in [0,160)
//   A[m, j] = xs[j%5][m*256 + (j/5)*8 + b],   W[j, o] = kernel[j*64 + o]
// One block per 16-node tile; 8 waves <-> 8 batches.
// LDS layout (single __shared__ array => offset 0):
//   [0,      40960)  sW : kernel weights [160][64]
//   [40960, 122880)  sA : xs slab [5][16][256]  (DMA'd in by the TDM)
// Wave 0 issues one 3D TENSOR_LOAD_TO_LDS covering the whole slab:
//   tile  = (x=256, y=16, z=5), tensor dims (256, M, 5),
//   strides: dim0 = 256 elems, dim1 = M*256 elems, data_size = 4B.
// ---------------------------------------------------------------------------
__global__ __launch_bounds__(256) void cheb_wmma(const float* __restrict__ xs,   // [K_][M_][C_]
                                                 const float* __restrict__ kern, // [KD_][FOUT_]
                                                 float* __restrict__ out) {      // [B_][M_][FOUT_]
    __shared__ float smem[SW_FLOATS + SA_FLOATS];      // 120 KB total
    float* sW = smem;
    float* sA = smem + SW_FLOATS;

    const int wave = threadIdx.x >> 5;      // 0..7 -> batch index b
    const int lane = threadIdx.x & 31;
    const int b    = wave;
    const int m16  = blockIdx.x * 16;
    const int rowA = lane & 15;             // A-matrix row for this lane
    const int hi   = lane >> 4;             // lane group: 0 -> K pair (0,1), 1 -> (2,3)

    // ---- TDM: DMA the 5x16x256 f32 slab of xs into LDS (wave 0 only) ----
    if (wave == 0) {
        const uint64_t ga = (uint64_t)(uintptr_t)xs + (uint64_t)m16 * (C_ * 4);
        // D# group 0: count=1 | lds_addr | global_addr[56:0] | type=2
        u32x4 g0 = { 1u,
                     (unsigned)SA_LDS_OFF,
                     (unsigned)(ga & 0xFFFFFFFFu),
                     (unsigned)((ga >> 32) & 0x01FFFFFFu) | 0x80000000u };
        // D# group 1 (bit-packed, see ISA 8.4):
        //  s0: wg_mask=0 | data_size=2(4B) | no barrier/iterate/pad
        //  s1: barrier_addr=0 | tensor_dim0 lo16 (256)            -> 0x01000000
        //  s2: tensor_dim0 hi16 (0) | tensor_dim1 lo16 (49152)    -> 0xC0000000
        //  s3: tensor_dim1 hi16 (0) | tile_dim0 (256)             -> 0x01000000
        //  s4: tile_dim1=16 | tile_dim2=5                         -> 0x00050010
        //  s5: tensor_dim0_stride lo32 = 256
        //  s6: tensor_dim0_stride hi16 (0) | tensor_dim1_stride lo16 (0)
        //  s7: tensor_dim1_stride[47:16] = (M*256)>>16 = 0xC0
        i32x8 g1 = { 0x00020000,
                     0x01000000,
                     (int)0xC0000000,
                     0x01000000,
                     0x00050010,
                     256,
                     0,
                     0xC0 };
        // D# group 2: tensor_dim2 = 5 (z length), rest unused for 3D tile
        i32x4 g2 = { K_, 0, 0, 0 };
        i32x4 g3 = { 0, 0, 0, 0 };
        // 6-arg toolchain variant: extra int32x8 group (zero-filled), cpol last
        i32x8 g4 = { 0, 0, 0, 0, 0, 0, 0, 0 };
        __builtin_amdgcn_tensor_load_to_lds(g0, g1, g2, g3, g4, 0);
    }

    // Stage the weight matrix while the TDM runs.
    for (int i = threadIdx.x; i < SW_FLOATS; i += 256) sW[i] = kern[i];

    if (wave == 0) {
        __builtin_amdgcn_s_wait_tensorcnt(0);   // slab landed in LDS
    }
    __syncthreads();                            // publish sA + sW to all waves

    v8f acc[4];
#pragma unroll
    for (int nt = 0; nt < 4; ++nt) acc[nt] = (v8f){0,0,0,0,0,0,0,0};

#pragma unroll 1
    for (int jc = 0; jc < KD_; jc += 4) {
        const int j0 = jc + hi * 2;
        const int j1 = j0 + 1;
        const int k0 = j0 % K_, f0 = j0 / K_;
        const int k1 = j1 % K_, f1 = j1 / K_;

        // A fragment from LDS: sA[k][rowA][f*8 + b]
        v2f a;
        a.x = sA[(k0 * 16 + rowA) * C_ + f0 * B_ + b];
        a.y = sA[(k1 * 16 + rowA) * C_ + f1 * B_ + b];

#pragma unroll
        for (int nt = 0; nt < 4; ++nt) {
            const int col = nt * 16 + (lane & 15);
            v2f bm;
            bm.x = sW[j0 * FOUT_ + col];
            bm.y = sW[j1 * FOUT_ + col];
            // 8 args: (neg_a, A, neg_b, B, c_mod, C, reuse_a, reuse_b)
            acc[nt] = __builtin_amdgcn_wmma_f32_16x16x4_f32(
                false, a, false, bm, (short)0, acc[nt], false, false);
        }
    }

    // C/D layout (16x16 f32): VGPR r -> lanes 0-15: (M=r, N=lane),
    //                                   lanes 16-31: (M=8+r, N=lane-16)
#pragma unroll
    for (int nt = 0; nt < 4; ++nt) {
#pragma unroll
        for (int r = 0; r < 8; ++r) {
            const int Mrow = r + hi * 8;
            const int Ncol = nt * 16 + (lane & 15);
            out[((size_t)b * M_ + m16 + Mrow) * FOUT_ + Ncol] = acc[nt][r];
        }
    }
}

// ---------------------------------------------------------------------------
extern "C" void kernel_launch(void* const* d_in, const int* in_sizes, int n_in,
                              void* d_out, int out_size, void* d_ws, size_t ws_size,
                              hipStream_t stream) {
    const float* x        = (const float*)d_in[0];
    // d_in[1] = lap_rows: implicit (rows sorted, DEG_ per row) -> unused
    const int*   lap_cols = (const int*)  d_in[2];
    const float* lap_vals = (const float*)d_in[3];
    const float* kern     = (const float*)d_in[4];
    float*       out      = (float*)d_out;

    // workspace: xs[k] each M_*C_ floats, k = 0..4  (5 * 48 MB)
    float* xs  = (float*)d_ws;
    const size_t plane = (size_t)M_ * C_;
    float* x0 = xs + 0 * plane;
    float* x1 = xs + 1 * plane;
    float* x2 = xs + 2 * plane;
    float* x3 = xs + 3 * plane;
    float* x4 = xs + 4 * plane;

    cheb_permute<<<(M_ * C_) / 256, 256, 0, stream>>>(x, x0);

    // x1 = L x0 ; x_k = 2 L x_{k-1} - x_{k-2}
    cheb_spmm<<<M_, 256, 0, stream>>>(lap_cols, lap_vals, x0, x0, x1, 1.0f,  0.0f);
    cheb_spmm<<<M_, 256, 0, stream>>>(lap_cols, lap_vals, x1, x0, x2, 2.0f, -1.0f);
    cheb_spmm<<<M_, 256, 0, stream>>>(lap_cols, lap_vals, x2, x1, x3, 2.0f, -1.0f);
    cheb_spmm<<<M_, 256, 0, stream>>>(lap_cols, lap_vals, x3, x2, x4, 2.0f, -1.0f);

    cheb_wmma<<<M_ / 16, 256, 0, stream>>>(xs, kern, out);
}